// AttentionEdgeModel_5420248727649
// MI455X (gfx1250) — compile-verified
//
#include <hip/hip_runtime.h>
#include <hip/hip_bf16.h>

// ---------------------------------------------------------------------------
// AttentionEdgeModel for MI455X (gfx1250, wave32).
//   p_src = x_s @ W_src^T, p_tgt = x_t @ W_tgt^T      (node projections, WMMA f32)
//   h[e]  = p_src[src] + p_tgt[tgt] + edge_attr @ W_edge^T   (edge pass, WMMA f32)
//   e     = leakyrelu(h . w_attn); scatter-softmax over src; RMSNorm * norm_w
// Memory-bound (~1 GB HBM traffic).  Streaming traffic (edge_attr, h, output)
// is marked non-temporal so the L2 keeps the 25.6 MB p_src/p_tgt gather tables
// (each row reused ~20x) and the 16 KB W_edge resident.
// ---------------------------------------------------------------------------

#define NSRC   50000
#define NEDGE  1000000
#define DD     64      // D_EDGE / output dim
#define DIN    128     // D_SRC / D_TGT

typedef __attribute__((ext_vector_type(2))) float v2f;
typedef __attribute__((ext_vector_type(4))) float v4f;
typedef __attribute__((ext_vector_type(8))) float v8f;

__device__ __forceinline__ v8f wmma4(v2f a, v2f b, v8f c) {
  // V_WMMA_F32_16X16X4_F32 : D = A(16x4) * B(4x16) + C(16x16), all f32
  return __builtin_amdgcn_wmma_f32_16x16x4_f32(
      false, a, false, b, (short)0, c, false, false);
}

// float atomic max via integer atomics (works for mixed signs; init = -inf)
__device__ __forceinline__ void atomicMaxF(float* addr, float val) {
  if (val >= 0.0f) atomicMax((int*)addr, __float_as_int(val));
  else             atomicMin((unsigned int*)addr, __float_as_uint(val));
}

__global__ void init_seg_kernel(float* __restrict__ m, float* __restrict__ den, int n) {
  int i = blockIdx.x * blockDim.x + threadIdx.x;
  if (i < n) { m[i] = -__builtin_inff(); den[i] = 0.0f; }
}

// P[nrows,64] = X[nrows,DIN] @ W[64,DIN]^T  -- one wave per 16-row tile
__global__ __launch_bounds__(256)
void proj_kernel(const float* __restrict__ X, const float* __restrict__ W,
                 float* __restrict__ P, int nrows) {
  const int lane = threadIdx.x & 31;
  const int wave = threadIdx.x >> 5;
  const int tile = blockIdx.x * 8 + wave;
  if (tile * 16 >= nrows) return;             // wave-uniform: EXEC stays all-1s
  const int l16  = lane & 15;
  const int hi   = lane >> 4;                 // 0: K=0,1 / rows 0-7 of C ; 1: K=2,3 / rows 8-15
  const int koff = hi * 2;
  const size_t arow = (size_t)(tile * 16 + l16);

  v8f c[4] = { {}, {}, {}, {} };
#pragma unroll
  for (int nb = 0; nb < 4; ++nb) {
    const size_t n = (size_t)(nb * 16 + l16);
#pragma unroll
    for (int ks = 0; ks < DIN / 4; ++ks) {
      // x rows are streamed once -> NT; W (32 KB) stays cached -> RT
      v2f a = __builtin_nontemporal_load((const v2f*)(X + arow * DIN + ks * 4 + koff));
      v2f b = *(const v2f*)(W + n * DIN + ks * 4 + koff);
      c[nb] = wmma4(a, b, c[nb]);
    }
  }
  // C layout: element r of c[nb] -> (row = r + hi*8, col = nb*16 + l16)
  // P is the gather table: store with default policy so it lands in L2.
#pragma unroll
  for (int nb = 0; nb < 4; ++nb) {
    const int col = nb * 16 + l16;
#pragma unroll
    for (int r = 0; r < 8; ++r)
      P[(size_t)(tile * 16 + r + hi * 8) * DD + col] = c[nb][r];
  }
}

// Edge pass 1: h = EA@We^T + gather(p_src) + gather(p_tgt); write h; e=leakyrelu(h.w_attn);
// segment max into Mbuf.  One wave per 16-edge tile.
__global__ __launch_bounds__(256)
void edge_pass1(const float* __restrict__ EA, const float* __restrict__ We,
                const float* __restrict__ Ps, const float* __restrict__ Pt,
                const int*   __restrict__ eidx, const float* __restrict__ w_attn,
                float* __restrict__ Hout, float* __restrict__ Ebuf,
                float* __restrict__ Mbuf) {
  const int lane = threadIdx.x & 31;
  const int wave = threadIdx.x >> 5;
  const int tile = blockIdx.x * 8 + wave;
  if (tile >= NEDGE / 16) return;             // wave-uniform
  const int l16  = lane & 15;
  const int hi   = lane >> 4;
  const int koff = hi * 2;
  const size_t arow = (size_t)(tile * 16 + l16);

  // Preload the 16 A-fragments of this wave's 16x64 edge_attr tile (streamed once -> NT)
  v2f a[16];
#pragma unroll
  for (int ks = 0; ks < 16; ++ks)
    a[ks] = __builtin_nontemporal_load((const v2f*)(EA + arow * DD + ks * 4 + koff));

  v8f c[4] = { {}, {}, {}, {} };
#pragma unroll
  for (int nb = 0; nb < 4; ++nb) {
    const size_t n = (size_t)(nb * 16 + l16);
#pragma unroll
    for (int ks = 0; ks < 16; ++ks) {
      v2f b = *(const v2f*)(We + n * DD + ks * 4 + koff);      // 16 KB, cache-hot
      c[nb] = wmma4(a[ks], b, c[nb]);
    }
  }

  // Gather node projections (tables are L2-resident, rows fully reused by the wave)
  int sid[8], tid[8];
#pragma unroll
  for (int r = 0; r < 8; ++r) {
    const int m = tile * 16 + r + hi * 8;
    sid[r] = eidx[m];
    tid[r] = eidx[NEDGE + m];
  }
#pragma unroll
  for (int nb = 0; nb < 4; ++nb) {
    const int col = nb * 16 + l16;
#pragma unroll
    for (int r = 0; r < 8; ++r)
      c[nb][r] += Ps[(size_t)sid[r] * DD + col] + Pt[(size_t)tid[r] * DD + col];
  }

  // Store h (256 MB, re-read only in pass 3 -> NT) and attention partials
  float wa[4];
#pragma unroll
  for (int nb = 0; nb < 4; ++nb) wa[nb] = w_attn[nb * 16 + l16];

  float ep[8];
#pragma unroll
  for (int r = 0; r < 8; ++r) {
    const size_t row = (size_t)(tile * 16 + r + hi * 8);
    float s = 0.0f;
#pragma unroll
    for (int nb = 0; nb < 4; ++nb) {
      __builtin_nontemporal_store(c[nb][r], Hout + row * DD + nb * 16 + l16);
      s += c[nb][r] * wa[nb];
    }
    ep[r] = s;
  }
  // Reduce across the 16 lanes of each half-wave (xor masks 1..8 never flip bit 4)
#pragma unroll
  for (int r = 0; r < 8; ++r) {
#pragma unroll
    for (int off = 1; off < 16; off <<= 1)
      ep[r] += __shfl_xor(ep[r], off, 32);
  }
  if (l16 == 0) {                 // lane 0 -> rows 0-7, lane 16 -> rows 8-15
#pragma unroll
    for (int r = 0; r < 8; ++r) {
      float e = ep[r];
      e = (e >= 0.0f) ? e : 0.2f * e;            // leaky relu, slope 0.2
      Ebuf[tile * 16 + r + hi * 8] = e;
      atomicMaxF(&Mbuf[sid[r]], e);
    }
  }
}

// Edge pass 2: ex = exp(e - m[src]); segment-sum into denom
__global__ void edge_pass2(const int* __restrict__ eidx, const float* __restrict__ Ebuf,
                           const float* __restrict__ Mbuf, float* __restrict__ EXbuf,
                           float* __restrict__ Dbuf) {
  int e = blockIdx.x * blockDim.x + threadIdx.x;
  if (e >= NEDGE) return;
  int s = eidx[e];
  float ex = __expf(Ebuf[e] - Mbuf[s]);
  EXbuf[e] = ex;
  atomicAdd(&Dbuf[s], ex);
}

// Edge pass 3: alpha = ex/denom[src]; h *= alpha; RMSNorm * norm_w.
// 16 lanes per edge (v4f each), 2 edges per wave -> coalesced 256B segments.
__global__ __launch_bounds__(256)
void edge_pass3(const int* __restrict__ eidx, const float* __restrict__ EXbuf,
                const float* __restrict__ Dbuf, const float* __restrict__ norm_w,
                float* __restrict__ Out) {
  const int lane = threadIdx.x & 31;
  const int wave = threadIdx.x >> 5;
  const int hi   = lane >> 4;
  const int sub  = lane & 15;
  const int edge = (blockIdx.x * 8 + wave) * 2 + hi;
  if (edge >= NEDGE) return;
  const float alpha = EXbuf[edge] / Dbuf[eidx[edge]];
  v4f h = __builtin_nontemporal_load((const v4f*)(Out + (size_t)edge * DD + sub * 4));
  h *= alpha;
  float ss = h.x * h.x + h.y * h.y + h.z * h.z + h.w * h.w;
#pragma unroll
  for (int off = 1; off < 16; off <<= 1) ss += __shfl_xor(ss, off, 32);
  const float rms = __frsqrt_rn(ss * (1.0f / 64.0f) + 1.1920929e-7f);
  v4f nw = *(const v4f*)(norm_w + sub * 4);
  h = h * (rms * nw);
  __builtin_nontemporal_store(h, (v4f*)(Out + (size_t)edge * DD + sub * 4));
}

extern "C" void kernel_launch(void* const* d_in, const int* in_sizes, int n_in,
                              void* d_out, int out_size, void* d_ws, size_t ws_size,
                              hipStream_t stream) {
  const float* x_s    = (const float*)d_in[0];
  const float* x_t    = (const float*)d_in[1];
  const int*   eidx   = (const int*)  d_in[2];   // [2, NEDGE]
  const float* eattr  = (const float*)d_in[3];   // [NEDGE, 64]
  /* d_in[4] = x_u : unused by the reference computation */
  const float* W_src  = (const float*)d_in[5];   // [64,128]
  const float* W_tgt  = (const float*)d_in[6];   // [64,128]
  const float* W_edge = (const float*)d_in[7];   // [64,64]
  const float* w_attn = (const float*)d_in[8];   // [64]
  const float* norm_w = (const float*)d_in[9];   // [64]
  float* out = (float*)d_out;                    // [NEDGE, 64]

  // workspace layout (~34 MB of floats)
  float* ws     = (float*)d_ws;
  float* p_src  = ws;                              // NSRC*64
  float* p_tgt  = p_src  + (size_t)NSRC * DD;      // NSRC*64
  float* e_buf  = p_tgt  + (size_t)NSRC * DD;      // NEDGE
  float* ex_buf = e_buf  + NEDGE;                  // NEDGE
  float* m_buf  = ex_buf + NEDGE;                  // NSRC
  float* d_buf  = m_buf  + NSRC;                   // NSRC

  init_seg_kernel<<<(NSRC + 255) / 256, 256, 0, stream>>>(m_buf, d_buf, NSRC);

  const int projTiles = NSRC / 16;                 // 3125
  const int projBlocks = (projTiles + 7) / 8;
  proj_kernel<<<projBlocks, 256, 0, stream>>>(x_s, W_src, p_src, NSRC);
  proj_kernel<<<projBlocks, 256, 0, stream>>>(x_t, W_tgt, p_tgt, NSRC);

  const int edgeTiles = NEDGE / 16;                // 62500
  edge_pass1<<<(edgeTiles + 7) / 8, 256, 0, stream>>>(
      eattr, W_edge, p_src, p_tgt, eidx, w_attn, out, e_buf, m_buf);

  edge_pass2<<<(NEDGE + 255) / 256, 256, 0, stream>>>(eidx, e_buf, m_buf, ex_buf, d_buf);

  edge_pass3<<<(edgeTiles + 7) / 8, 256, 0, stream>>>(eidx, ex_buf, d_buf, norm_w, out);
}